// R_HGT_8959301780058
// MI455X (gfx1250) — compile-verified
//
#include <hip/hip_runtime.h>
#include <math.h>

// ---------------- problem constants (match reference) ----------------
#define NN 100000      // nodes
#define EE 600000      // edges per relation
#define RR 2           // relations
#define HH 8           // heads
#define DHD 16         // per-head dim
#define DD 128         // H*DH = hidden
#define RIN 64         // relation input dim
#define SLOPE 0.2f

#define BT_STRIDE 132  // padded LDS row stride (floats): bank = (4*l + k) % 64, conflict-free

typedef __attribute__((ext_vector_type(2))) float v2f;
typedef __attribute__((ext_vector_type(8))) float v8f;

__device__ __forceinline__ float leaky(float v) { return v >= 0.f ? v : SLOPE * v; }

// order-preserving float -> uint map (for atomic segment-max over any-sign floats)
__device__ __forceinline__ unsigned ford(float f) {
    unsigned b = __float_as_uint(f);
    return (b & 0x80000000u) ? ~b : (b | 0x80000000u);
}
__device__ __forceinline__ float funord(unsigned u) {
    unsigned b = (u & 0x80000000u) ? (u & 0x7FFFFFFFu) : ~u;
    return __uint_as_float(b);
}

// Low 32 bits of a generic pointer to LDS == wave-relative LDS byte address
// (ISA §10.2 aperture mapping).  The ptrtoint also
// escapes the __shared__ object so the "memory"-clobbering asm below is
// treated as a potential writer and the ds_loads are NOT folded to undef.
__device__ __forceinline__ unsigned lds_addr_of(const void* p) {
    return (unsigned)(unsigned long long)p;
}

// ---------------- f32 WMMA GEMM: C[M,128] = A[M,128] @ W[128,128] + bias ----------------
// 4 waves / block, 8 M-tiles / block, one shared 16-col N-tile.
// B panel (128x16 = 8KB) staged once per block into LDS *transposed* via
// CDNA5 async-to-LDS copies (ASYNCcnt path), then each wave runs two
// independent WMMA chains (2 M-tiles) sharing one ds_load_b64 B fragment/step.
// WMMA f32 16x16x4 layouts:
//   A: lanes 0-15 row M=lane K={k,k+1}; lanes 16-31 row M=lane-16 K={k+2,k+3}
//   B: mirrored per-column.  C/D: vgpr j -> row j + 8*(lane>=16), col lane&15.
__global__ __launch_bounds__(128) void gemm128_wmma(const float* __restrict__ A,
                                                    const float* __restrict__ W,
                                                    const float* __restrict__ bias,
                                                    float* __restrict__ C) {
    __shared__ float bt[16 * BT_STRIDE];

    const int t     = threadIdx.x;
    const int tileN = blockIdx.y << 4;
    const size_t blockM = (size_t)blockIdx.x * 128;   // 8 M-tiles per block

    // ---- stage B panel transposed into LDS with async copies (transpose = per-lane scatter)
    {
        const int j  = t & 15;   // column within N-tile
        const int kb = t >> 4;   // 0..7
        const float* gsrc = W + (size_t)kb * DD + tileN + j;
        const unsigned lbase = lds_addr_of(&bt[j * BT_STRIDE + kb]);
#pragma unroll
        for (int i = 0; i < 16; ++i) {                 // op i covers K row kb + 8*i
            unsigned long long ga = (unsigned long long)(gsrc + (size_t)i * 8 * DD);
            unsigned la = lbase + (unsigned)(i * 8 * 4);
            asm volatile("global_load_async_to_lds_b32 %0, %1, off"
                         :: "v"(la), "v"(ga) : "memory");
        }
        asm volatile("s_wait_asynccnt 0" ::: "memory");
    }
    __syncthreads();

    const int wave = t >> 5;
    const int lane = t & 31;
    const int half = lane >> 4;
    const int l    = lane & 31 & 15;

    const size_t m0 = blockM + (size_t)wave * 32;      // first M-tile of this wave
    const size_t m1 = m0 + 16;                         // second M-tile
    const bool d0 = m0 < (size_t)NN;
    const bool d1 = m1 < (size_t)NN;
    if (!d0) return;                                   // idle tail wave (after barrier)

    const float* a0p = A + (m0 + l) * DD + (half << 1);
    const float* a1p = A + ((d1 ? m1 : m0) + l) * DD + (half << 1);   // clamp OOB reads
    const float* bp_ = &bt[l * BT_STRIDE + (half << 1)];

    v8f acc0 = {0.f, 0.f, 0.f, 0.f, 0.f, 0.f, 0.f, 0.f};
    v8f acc1 = acc0;
#pragma unroll
    for (int k = 0; k < DD; k += 4) {
        v2f b  = *reinterpret_cast<const v2f*>(bp_ + k);   // ds_load_b64, conflict-free
        v2f a0 = *reinterpret_cast<const v2f*>(a0p + k);   // global_load_b64
        v2f a1 = *reinterpret_cast<const v2f*>(a1p + k);
        acc0 = __builtin_amdgcn_wmma_f32_16x16x4_f32(false, a0, false, b,
                                                     (short)0, acc0, false, false);
        acc1 = __builtin_amdgcn_wmma_f32_16x16x4_f32(false, a1, false, b,
                                                     (short)0, acc1, false, false);
    }
    const float bv = bias[tileN + l];
#pragma unroll
    for (int j = 0; j < 8; ++j)
        C[(m0 + j + (half << 3)) * DD + tileN + l] = acc0[j] + bv;
    if (d1) {
#pragma unroll
        for (int j = 0; j < 8; ++j)
            C[(m1 + j + (half << 3)) * DD + tileN + l] = acc1[j] + bv;
    }
}

// ---------------- relation attention vectors: a[r] = rel_emb[r] @ Wrel[r]  ([H,2,DH]=256) ----
__global__ void rel_attn_vec(const float* __restrict__ rel_emb,
                             const float* __restrict__ Wrel,
                             float* __restrict__ a) {
    int idx = blockIdx.x * blockDim.x + threadIdx.x;
    if (idx >= RR * 256) return;
    int r = idx >> 8, o = idx & 255;
    const float* e = rel_emb + r * RIN;
    const float* w = Wrel + (size_t)r * RIN * 256 + o;
    float acc = 0.f;
#pragma unroll 8
    for (int i = 0; i < RIN; ++i) acc += e[i] * w[(size_t)i * 256];
    a[idx] = acc;
}

// ---------------- per-(r,n,h) attention scores -------------------------------------------
__global__ void node_scores(const float* __restrict__ s, const float* __restrict__ hb,
                            const float* __restrict__ a,
                            float* __restrict__ sc_src, float* __restrict__ sc_dst) {
    int idx = blockIdx.x * blockDim.x + threadIdx.x;  // [R,N,H]
    if (idx >= RR * NN * HH) return;
    int h = idx % HH;
    int n = (idx / HH) % NN;
    int r = idx / (NN * HH);
    const float* sv = s + ((size_t)r * NN + n) * DD + h * DHD;
    const float* hv = hb + (size_t)n * DD + h * DHD;
    const float* a0 = a + r * 256 + h * 32;        // a[r,h,0,:]
    const float* a1 = a0 + DHD;                    // a[r,h,1,:]
    float ss = 0.f, sd = 0.f;
#pragma unroll
    for (int d = 0; d < DHD; ++d) { ss += sv[d] * a0[d]; sd += hv[d] * a1[d]; }
    sc_src[idx] = ss;
    sc_dst[idx] = sd;
}

// ---------------- edge softmax passes (atomics) ------------------------------------------
__global__ void edge_max(const int* __restrict__ src, const int* __restrict__ dst,
                         const float* __restrict__ sc_src, const float* __restrict__ sc_dst,
                         unsigned* __restrict__ maxbuf) {
    size_t idx = (size_t)blockIdx.x * blockDim.x + threadIdx.x;  // [R,E,H]
    if (idx >= (size_t)RR * EE * HH) return;
    int h = idx % HH;
    size_t re = idx / HH;
    int e = re % EE, r = re / EE;
    int sr = src[(size_t)r * EE + e], dr = dst[(size_t)r * EE + e];
    float v = leaky(sc_src[((size_t)r * NN + sr) * HH + h] +
                    sc_dst[((size_t)r * NN + dr) * HH + h]);
    atomicMax(&maxbuf[((size_t)r * NN + dr) * HH + h], ford(v));
}

__global__ void edge_expsum(const int* __restrict__ src, const int* __restrict__ dst,
                            const float* __restrict__ sc_src, const float* __restrict__ sc_dst,
                            const unsigned* __restrict__ maxbuf, float* __restrict__ sumbuf) {
    size_t idx = (size_t)blockIdx.x * blockDim.x + threadIdx.x;
    if (idx >= (size_t)RR * EE * HH) return;
    int h = idx % HH;
    size_t re = idx / HH;
    int e = re % EE, r = re / EE;
    int sr = src[(size_t)r * EE + e], dr = dst[(size_t)r * EE + e];
    size_t di = ((size_t)r * NN + dr) * HH + h;
    float v = leaky(sc_src[((size_t)r * NN + sr) * HH + h] + sc_dst[di]);
    float ex = __expf(v - funord(maxbuf[di]));
    atomicAdd(&sumbuf[di], ex);
}

__global__ void edge_scatter(const int* __restrict__ src, const int* __restrict__ dst,
                             const float* __restrict__ sc_src, const float* __restrict__ sc_dst,
                             const unsigned* __restrict__ maxbuf, const float* __restrict__ sumbuf,
                             const float* __restrict__ s, float* __restrict__ outbuf) {
    size_t idx = (size_t)blockIdx.x * blockDim.x + threadIdx.x;
    if (idx >= (size_t)RR * EE * HH) return;
    int h = idx % HH;
    size_t re = idx / HH;
    int e = re % EE, r = re / EE;
    int sr = src[(size_t)r * EE + e], dr = dst[(size_t)r * EE + e];
    size_t di = ((size_t)r * NN + dr) * HH + h;
    float v = leaky(sc_src[((size_t)r * NN + sr) * HH + h] + sc_dst[di]);
    float alpha = __expf(v - funord(maxbuf[di])) / sumbuf[di];
    const float4* sv4 = reinterpret_cast<const float4*>(
        s + ((size_t)r * NN + sr) * DD + h * DHD);          // 64B-aligned -> b128 loads
    float* ov = outbuf + ((size_t)r * NN + dr) * DD + h * DHD;
#pragma unroll
    for (int q = 0; q < 4; ++q) {
        float4 m4 = sv4[q];
        atomicAdd(&ov[q * 4 + 0], alpha * m4.x);
        atomicAdd(&ov[q * 4 + 1], alpha * m4.y);
        atomicAdd(&ov[q * 4 + 2], alpha * m4.z);
        atomicAdd(&ov[q * 4 + 3], alpha * m4.w);
    }
}

// ---------------- gated residual + relation crossing -> d_out ----------------------------
__global__ void gate_cross(const float* __restrict__ outbuf, const float* __restrict__ res,
                           const float* __restrict__ res_alpha, const float* __restrict__ crossW,
                           float* __restrict__ fused) {
    int idx = blockIdx.x * blockDim.x + threadIdx.x;  // [N,H]
    if (idx >= NN * HH) return;
    int h = idx % HH, n = idx / HH;
    float g = 1.f / (1.f + __expf(-res_alpha[0]));
    float o[RR][DHD];
    const float* rv = res + (size_t)n * DD + h * DHD;
#pragma unroll
    for (int r = 0; r < RR; ++r) {
        const float* ov = outbuf + ((size_t)r * NN + n) * DD + h * DHD;
#pragma unroll
        for (int d = 0; d < DHD; ++d) o[r][d] = ov[d] * g + rv[d] * (1.f - g);
    }
#pragma unroll
    for (int r = 0; r < RR; ++r) {
        const float* cw = crossW + (r * HH + h) * DHD;
        float l0 = 0.f, l1 = 0.f;
#pragma unroll
        for (int d = 0; d < DHD; ++d) { l0 += o[0][d] * cw[d]; l1 += o[1][d] * cw[d]; }
        l0 = leaky(l0); l1 = leaky(l1);
        float m  = fmaxf(l0, l1);
        float e0 = __expf(l0 - m), e1 = __expf(l1 - m);
        float inv = 1.f / (e0 + e1);
        float a0 = e0 * inv, a1 = e1 * inv;
        float* dp = fused + ((size_t)r * NN + n) * DD + h * DHD;
#pragma unroll
        for (int d = 0; d < DHD; ++d) dp[d] = a0 * o[0][d] + a1 * o[1][d];
    }
}

// ---------------- relation propagation: rel_out[r] = rel_emb[r]@Wprop[r] + bprop[r] ------
__global__ void rel_prop(const float* __restrict__ rel_emb, const float* __restrict__ Wprop,
                         const float* __restrict__ bprop, float* __restrict__ out) {
    int idx = blockIdx.x * blockDim.x + threadIdx.x;  // [R, 128]
    if (idx >= RR * 128) return;
    int r = idx >> 7, o = idx & 127;
    const float* e = rel_emb + r * RIN;
    const float* w = Wprop + (size_t)r * RIN * 128 + o;
    float acc = bprop[idx];
#pragma unroll 8
    for (int i = 0; i < RIN; ++i) acc += e[i] * w[(size_t)i * 128];
    out[idx] = acc;
}

// ---------------- launcher ----------------------------------------------------------------
extern "C" void kernel_launch(void* const* d_in, const int* in_sizes, int n_in,
                              void* d_out, int out_size, void* d_ws, size_t ws_size,
                              hipStream_t stream) {
    const float* x        = (const float*)d_in[0];
    const int*   src      = (const int*)d_in[1];
    const int*   dst      = (const int*)d_in[2];
    const float* rel_emb  = (const float*)d_in[3];
    const float* Wp       = (const float*)d_in[4];
    const float* bp       = (const float*)d_in[5];
    const float* Wn       = (const float*)d_in[6];
    const float* bn       = (const float*)d_in[7];
    const float* Wsrc     = (const float*)d_in[8];
    const float* bsrc     = (const float*)d_in[9];
    const float* Wrel     = (const float*)d_in[10];
    const float* resW     = (const float*)d_in[11];
    const float* resb     = (const float*)d_in[12];
    const float* res_alpha= (const float*)d_in[13];
    const float* crossW   = (const float*)d_in[14];
    const float* Wprop    = (const float*)d_in[15];
    const float* bprop    = (const float*)d_in[16];
    float* out = (float*)d_out;

    const size_t ND  = (size_t)NN * DD;     // 12.8M floats
    const size_t RNH = (size_t)RR * NN * HH;

    float* ws     = (float*)d_ws;
    float* xp     = ws;                 // [N,D]      (dead after node_scores)
    float* hb     = ws + ND;            // [N,D]      (dead after node_scores)
    float* outbuf = ws;                 // [R,N,D]    aliases xp+hb once they are dead
    float* res    = ws + 2 * ND;        // [N,D]
    float* sbuf   = ws + 3 * ND;        // [R,N,D]
    float* abuf   = ws + 5 * ND;        // 512
    float* scsrc  = abuf + 512;         // [R,N,H]
    float* scdst  = scsrc + RNH;        // [R,N,H]
    unsigned* maxb= (unsigned*)(scdst + RNH);  // [R,N,H]
    float* sumb   = scdst + 2 * RNH;    // [R,N,H]

    const unsigned mblocks = (unsigned)((NN / 16 + 7) / 8);   // 8 M-tiles per block
    const dim3 gemmGrid(mblocks, DD / 16);
    // Phase A: dense GEMMs on the WMMA pipe
    gemm128_wmma<<<gemmGrid, 128, 0, stream>>>(x,  Wp,   bp,   xp);
    gemm128_wmma<<<gemmGrid, 128, 0, stream>>>(xp, Wn,   bn,   hb);
    gemm128_wmma<<<gemmGrid, 128, 0, stream>>>(xp, resW, resb, res);
    gemm128_wmma<<<gemmGrid, 128, 0, stream>>>(hb, Wsrc,          bsrc,      sbuf);
    gemm128_wmma<<<gemmGrid, 128, 0, stream>>>(hb, Wsrc + DD*DD,  bsrc + DD, sbuf + ND);

    // Phase B: attention vectors + per-node scores
    rel_attn_vec<<<(RR * 256 + 255) / 256, 256, 0, stream>>>(rel_emb, Wrel, abuf);
    node_scores<<<(unsigned)((RNH + 255) / 256), 256, 0, stream>>>(sbuf, hb, abuf, scsrc, scdst);

    // Phase C: edge softmax (xp/hb are now dead -> outbuf may overwrite them)
    hipMemsetAsync(maxb,   0, RNH * sizeof(unsigned), stream);   // 0 < ford(v) for all v
    hipMemsetAsync(sumb,   0, RNH * sizeof(float), stream);
    hipMemsetAsync(outbuf, 0, 2 * ND * sizeof(float), stream);
    const size_t nedge = (size_t)RR * EE * HH;
    const unsigned eblocks = (unsigned)((nedge + 255) / 256);
    edge_max    <<<eblocks, 256, 0, stream>>>(src, dst, scsrc, scdst, maxb);
    edge_expsum <<<eblocks, 256, 0, stream>>>(src, dst, scsrc, scdst, maxb, sumb);
    edge_scatter<<<eblocks, 256, 0, stream>>>(src, dst, scsrc, scdst, maxb, sumb, sbuf, outbuf);

    // Phase D: gated residual + relation crossing -> d_out[0 : R*N*D]
    gate_cross<<<(NN * HH + 255) / 256, 256, 0, stream>>>(outbuf, res, res_alpha, crossW, out);

    // Phase E: relation propagation -> d_out[R*N*D : +R*128]
    rel_prop<<<1, 256, 0, stream>>>(rel_emb, Wprop, bprop, out + (size_t)RR * ND);
}